// Symmetric_nuLSQ_quantizer_52029233823754
// MI455X (gfx1250) — compile-verified
//
#include <hip/hip_runtime.h>

// Symmetric nuLSQ quantizer for MI455X (gfx1250).
// Memory-bound streaming op: 411 MB traffic -> ~17.6 us floor @ 23.3 TB/s.
// Data path: global_load_async_to_lds_b128 (ASYNCcnt, no in-flight VGPRs)
//            -> s_wait_asynccnt 0 -> ds_load_b128 -> 15-level branchless
//            quantize (SGPR LUT, VOPD-friendly) -> NT global_store_b128.
// No matrix structure -> no WMMA by design.

typedef float v4f __attribute__((ext_vector_type(4)));

#define QN   8
#define QP   7
#define NLEV (QN + QP)   // 15 thresholds -> 16 levels

__global__ __launch_bounds__(256)
void nulsq_quant_kernel(const float* __restrict__ x,
                        const float* __restrict__ scale,
                        float* __restrict__ y,
                        int n4, int n)
{
    // 16 KB LDS staging: one v4f slot per (thread, j). Each thread reads only
    // the slots it async-loaded itself, so no workgroup barrier is required.
    __shared__ v4f sbuf[256 * 4];

    // ---- uniform threshold/value setup (scalar loads -> SGPRs, s_add_f32) ----
    // Reference rounding order preserved:
    //   c = cumsum(scale);  thr = c - 0.5f*s;  level partial sums in ascending k.
    float T[NLEV], V[NLEV];
    {
        float c = 0.0f;
#pragma unroll
        for (int k = 0; k < QN; ++k) {            // negative side: scale[0..7]
            float s = scale[k];                   // uniform -> s_load
            c = c + s;                            // cneg[k]
            T[k] = -(c - 0.5f * s);               // x <  T[k]  <=>  -x > cneg[k]-0.5*s (strict)
            V[k] = -s;
        }
        c = 0.0f;
#pragma unroll
        for (int k = 0; k < QP; ++k) {            // positive side: scale[8..14]
            float s = scale[QN + k];
            c = c + s;                            // cpos[k]
            T[QN + k] = c - 0.5f * s;             // x > T[k] (strict, searchsorted 'left')
            V[QN + k] = s;
        }
    }

    // Branchless 4-wide quantize; 4 independent accumulator chains for
    // VOPD dual-issue. Ascending-k adds reproduce cumsum rounding exactly.
    auto quant4 = [&](v4f xx) -> v4f {
        v4f acc = {0.0f, 0.0f, 0.0f, 0.0f};
#pragma unroll
        for (int k = 0; k < QN; ++k) {
#pragma unroll
            for (int e = 0; e < 4; ++e)
                acc[e] += (xx[e] < T[k]) ? V[k] : 0.0f;
        }
#pragma unroll
        for (int k = QN; k < NLEV; ++k) {
#pragma unroll
            for (int e = 0; e < 4; ++e)
                acc[e] += (xx[e] > T[k]) ? V[k] : 0.0f;
        }
        return acc;
    };

    auto quant1 = [&](float xx) -> float {
        float acc = 0.0f;
#pragma unroll
        for (int k = 0; k < QN; ++k)    acc += (xx < T[k]) ? V[k] : 0.0f;
#pragma unroll
        for (int k = QN; k < NLEV; ++k) acc += (xx > T[k]) ? V[k] : 0.0f;
        return acc;
    };

    const int tid  = threadIdx.x;
    const int base = blockIdx.x * (256 * 4) + tid;   // 4 float4 tiles / thread

    const v4f* __restrict__ xv = (const v4f*)x;
    v4f*       __restrict__ yv = (v4f*)y;

    // Streaming prefetch ~256 KB ahead of this block's region (global_prefetch_b8).
    if (base + 16384 < n4)
        __builtin_prefetch((const void*)(xv + base + 16384), 0, 1);

    if (base + 3 * 256 < n4) {
        // ---- async stage HBM -> LDS (ASYNCcnt path), self-slot per lane ----
#pragma unroll
        for (int j = 0; j < 4; ++j) {
            // Flat LDS aperture keeps the LDS byte offset in the low 32 bits.
            unsigned lds_off = (unsigned)(uintptr_t)(&sbuf[tid + 256 * j]);
            const v4f* g = xv + (base + 256 * j);
            asm volatile("global_load_async_to_lds_b128 %0, %1, off"
                         :
                         : "v"(lds_off), "v"(g)
                         : "memory");
        }
        asm volatile("s_wait_asynccnt 0" ::: "memory");

        v4f a0 = sbuf[tid];
        v4f a1 = sbuf[tid + 256];
        v4f a2 = sbuf[tid + 512];
        v4f a3 = sbuf[tid + 768];

        v4f r0 = quant4(a0);
        v4f r1 = quant4(a1);
        v4f r2 = quant4(a2);
        v4f r3 = quant4(a3);

        __builtin_nontemporal_store(r0, yv + base);
        __builtin_nontemporal_store(r1, yv + base + 256);
        __builtin_nontemporal_store(r2, yv + base + 512);
        __builtin_nontemporal_store(r3, yv + base + 768);
    } else {
        // Boundary block: guarded direct NT loads (async only on full tiles).
#pragma unroll
        for (int j = 0; j < 4; ++j) {
            int idx = base + 256 * j;
            if (idx < n4) {
                v4f a = __builtin_nontemporal_load(xv + idx);
                __builtin_nontemporal_store(quant4(a), yv + idx);
            }
        }
    }

    // Scalar tail (n % 4 elements); n divides by 4 for the given shape but
    // stay generic. rem < 4 so only lanes 0..2 of block 0 fire.
    int rem = n - 4 * n4;
    if (rem > 0) {
        int gid = blockIdx.x * 256 + tid;
        if (blockIdx.x == 0 && gid < rem) {
            int e = 4 * n4 + gid;
            y[e] = quant1(x[e]);
        }
    }
}

extern "C" void kernel_launch(void* const* d_in, const int* in_sizes, int n_in,
                              void* d_out, int out_size, void* d_ws, size_t ws_size,
                              hipStream_t stream) {
    const float* x     = (const float*)d_in[0];   // (64,256,56,56) f32
    const float* scale = (const float*)d_in[1];   // (15,) f32
    float*       y     = (float*)d_out;           // same shape as x, f32

    int n  = in_sizes[0];          // 51,380,224
    int n4 = n / 4;                // full float4 vectors
    int blocks = (n4 + 1023) / 1024;
    if (blocks < 1) blocks = 1;

    nulsq_quant_kernel<<<blocks, 256, 0, stream>>>(x, scale, y, n4, n);
}